// SequenceFoam_5703716569467
// MI455X (gfx1250) — compile-verified
//
#include <hip/hip_runtime.h>
#include <math.h>

// ---------------------------------------------------------------------------
// SequenceFoam, reduced analytically:
//   s_t = 1 (x) m_t + c3*B0,   B0 = bubbles - mean(bubbles)  (fixed)
//   m_t = c1*bmean + c2*(x_t + decay*mmean_{t-1})
//   mmean_t = decay*mmean_{t-1} + (1-decay)*m_t
//   s^T s = c3^2 * B0^T B0 + 64 * m m^T      (since 1^T B0 = 0)
// Eigenvalues: one-time eigensolve of W = B0 B0^T (64x64, rank 63),
// then per-step diag+rank1 secular equation in the 64-dim reduced space,
// plus 192 exact zero eigenvalues (-> EPS after clipping).
// ---------------------------------------------------------------------------

typedef __attribute__((ext_vector_type(2))) float v2f;
typedef __attribute__((ext_vector_type(8))) float v8f;

#define SEQ 1024
#define DD  256
#define NB  64
#define ETA 0.1f
#define EPSV 1e-10f

// workspace layout (float offsets)
constexpr size_t OFF_CONSTS = 0;        // 16   [0]=decay [1]=c1 [2]=c2 [3]=c3
constexpr size_t OFF_BMEAN  = 16;       // 256
constexpr size_t OFF_B0     = 272;      // 64*256
constexpr size_t OFF_W      = 16656;    // 64*64
constexpr size_t OFF_A2     = 20752;    // 64*64  A2 = diag(mu^-1/2) U^T (row0 = 0)
constexpr size_t OFF_MU     = 24848;    // 64
constexpr size_t OFF_D     = 24912;    // 64    d_i = c3^2 mu_i / 64 (sorted, d0=0)
constexpr size_t OFF_MSQ    = 24976;    // 1024  |m_t|^2
constexpr size_t OFF_M      = 26000;    // 256*1024  M[d][t]
constexpr size_t OFF_Y      = 288144;   // 64*1024   Y = B0 * M
constexpr size_t OFF_Z      = 353680;   // 64*1024   Z = A2 * Y
// total ~1.7 MB of workspace

// --------------------------- setup -----------------------------------------
__global__ __launch_bounds__(256) void k_setup(const float* __restrict__ bubbles,
                                               const float* __restrict__ mdecay,
                                               float* __restrict__ ws) {
  int d = threadIdx.x;
  float s = 0.f;
  for (int i = 0; i < NB; ++i) s += bubbles[i * DD + d];
  float bm = s * (1.0f / NB);
  ws[OFF_BMEAN + d] = bm;
  for (int i = 0; i < NB; ++i) ws[OFF_B0 + i * DD + d] = bubbles[i * DD + d] - bm;
  if (d == 0) {
    float decay = 1.0f / (1.0f + expf(-mdecay[0]));
    float a = 1.0f - ETA, c1 = 1.0f;
    float g = 1.0f - 2.0f * ETA, c3 = 1.0f;
    for (int k = 0; k < 5; ++k) { c1 *= a; c3 *= g; }
    ws[OFF_CONSTS + 0] = decay;
    ws[OFF_CONSTS + 1] = c1;
    ws[OFF_CONSTS + 2] = 1.0f - c1;
    ws[OFF_CONSTS + 3] = c3;
  }
}

// --------------------------- W = B0 * B0^T (64x64) --------------------------
__global__ __launch_bounds__(256) void k_gram(float* __restrict__ ws) {
  int idx = blockIdx.x * 256 + threadIdx.x;   // 4096 entries
  int p = idx >> 6, q = idx & 63;
  const float* B0 = ws + OFF_B0;
  float s = 0.f;
  for (int k = 0; k < DD; ++k) s += B0[p * DD + k] * B0[q * DD + k];
  ws[OFF_W + idx] = s;
}

// --------------------------- one-time 64x64 Jacobi eigensolve ---------------
__global__ __launch_bounds__(256) void k_jacobi(float* __restrict__ ws) {
  __shared__ float Ws[NB][NB];
  __shared__ float Us[NB][NB];
  __shared__ float cs[32], sn[32];
  __shared__ int perm[NB];
  int tid = threadIdx.x;
  for (int i = tid; i < NB * NB; i += 256) {
    Ws[i >> 6][i & 63] = ws[OFF_W + i];
    Us[i >> 6][i & 63] = ((i >> 6) == (i & 63)) ? 1.f : 0.f;
  }
  __syncthreads();
  for (int sweep = 0; sweep < 20; ++sweep) {
    for (int r = 0; r < 63; ++r) {
      // round-robin tournament: 32 disjoint (p,q) pairs per round
      if (tid < 32) {
        int p, q;
        if (tid == 0) { p = 63; q = r; }
        else { p = (r + tid) % 63; q = (r + 63 - tid) % 63; }
        float apq = Ws[p][q];
        float c = 1.f, s = 0.f;
        if (fabsf(apq) > 1e-13f) {
          float th = (Ws[q][q] - Ws[p][p]) / (2.f * apq);
          float t = (th >= 0.f ? 1.f : -1.f) / (fabsf(th) + sqrtf(th * th + 1.f));
          c = 1.0f / sqrtf(1.f + t * t);
          s = t * c;
        }
        cs[tid] = c; sn[tid] = s;
      }
      __syncthreads();
      // row rotations (J^T * W): pairs are row-disjoint
      for (int task = tid; task < 32 * NB; task += 256) {
        int pr = task >> 6, k = task & 63;
        int p, q;
        if (pr == 0) { p = 63; q = r; }
        else { p = (r + pr) % 63; q = (r + 63 - pr) % 63; }
        float c = cs[pr], s = sn[pr];
        float wp = Ws[p][k], wq = Ws[q][k];
        Ws[p][k] = c * wp - s * wq;
        Ws[q][k] = s * wp + c * wq;
      }
      __syncthreads();
      // column rotations (W * J) and eigenvector accumulation (U * J)
      for (int task = tid; task < 32 * NB; task += 256) {
        int pr = task >> 6, k = task & 63;
        int p, q;
        if (pr == 0) { p = 63; q = r; }
        else { p = (r + pr) % 63; q = (r + 63 - pr) % 63; }
        float c = cs[pr], s = sn[pr];
        float wp = Ws[k][p], wq = Ws[k][q];
        Ws[k][p] = c * wp - s * wq;
        Ws[k][q] = s * wp + c * wq;
        float up = Us[k][p], uq = Us[k][q];
        Us[k][p] = c * up - s * uq;
        Us[k][q] = s * up + c * uq;
      }
      __syncthreads();
    }
  }
  // sort eigenvalues ascending (64 values: trivial selection sort)
  if (tid == 0) {
    for (int i = 0; i < NB; ++i) perm[i] = i;
    for (int i = 0; i < NB; ++i) {
      int best = i;
      for (int j = i + 1; j < NB; ++j)
        if (Ws[perm[j]][perm[j]] < Ws[perm[best]][perm[best]]) best = j;
      int t = perm[i]; perm[i] = perm[best]; perm[best] = t;
    }
    float c3 = ws[OFF_CONSTS + 3];
    float f = c3 * c3 / (float)NB;
    float prev = 0.f;
    for (int i = 0; i < NB; ++i) {
      float mu = Ws[perm[i]][perm[i]];
      ws[OFF_MU + i] = mu;
      float di = (i == 0) ? 0.f : f * mu;       // index 0 = exact null (ones dir)
      if (di < prev) di = prev;                 // enforce nondecreasing / >= 0
      ws[OFF_D + i] = di;
      prev = di;
    }
  }
  __syncthreads();
  // A2[i][j] = U[j][perm[i]] / sqrt(mu_i), row 0 zeroed (null direction)
  for (int task = tid; task < NB * NB; task += 256) {
    int i = task >> 6, j = task & 63;
    int pi = perm[i];
    float mu = Ws[pi][pi];
    float a = 0.f;
    if (i > 0 && mu > 1e-20f) a = Us[j][pi] / sqrtf(mu);
    ws[OFF_A2 + i * NB + j] = a;
  }
}

// --------------------------- sequential scan for m_t ------------------------
__global__ __launch_bounds__(256) void k_scan(const int* __restrict__ tokens,
                                              const float* __restrict__ embed,
                                              float* __restrict__ ws) {
  __shared__ float red[8];
  int d = threadIdx.x;
  float decay = ws[OFF_CONSTS + 0];
  float c1 = ws[OFF_CONSTS + 1];
  float c2 = ws[OFF_CONSTS + 2];
  float bm = ws[OFF_BMEAN + d];
  float mm = 0.f;                               // mmean element (register-resident)
  int lane = d & 31, wid = d >> 5;
  for (int t = 0; t < SEQ; ++t) {
    if (t + 1 < SEQ)                            // gfx1250 global_prefetch_b8
      __builtin_prefetch(&embed[(size_t)tokens[t + 1] * DD + d], 0, 1);
    float x = embed[(size_t)tokens[t] * DD + d];
    float m = c1 * bm + c2 * (x + decay * mm);
    ws[OFF_M + (size_t)d * SEQ + t] = m;
    mm = decay * mm + (1.f - decay) * m;
    // |m|^2 via wave32 shuffle reduce + cross-wave LDS
    float v = m * m;
    for (int o = 16; o > 0; o >>= 1) v += __shfl_xor(v, o, 32);
    if (lane == 0) red[wid] = v;
    __syncthreads();
    if (d == 0) {
      float s = 0.f;
      for (int w = 0; w < 8; ++w) s += red[w];
      ws[OFF_MSQ + t] = s;
    }
    __syncthreads();
  }
}

// --------------------------- f32 WMMA GEMM: C(64x1024) = A(64xK) * B(Kx1024) -
template <int K, int LDA>
__global__ __launch_bounds__(256) void k_gemm_wmma(const float* __restrict__ A,
                                                   const float* __restrict__ B,
                                                   float* __restrict__ C) {
  int wid = blockIdx.x * (blockDim.x >> 5) + (threadIdx.x >> 5); // 256 waves
  int lane = threadIdx.x & 31;
  int tm = (wid & 3) * 16;          // 4 row tiles
  int tn = (wid >> 2) * 16;         // 64 col tiles
  int half = lane >> 4;             // 0: K 0..1 / 1: K 2..3 per ISA A/B layout
  int l = lane & 15;
  v8f acc = {};
#pragma unroll 4
  for (int k = 0; k < K; k += 4) {
    int ka = k + half * 2;
    v2f a, b;
    a.x = A[(tm + l) * LDA + ka];       // A 16x4 tile: lane row, 2 K per VGPR
    a.y = A[(tm + l) * LDA + ka + 1];
    b.x = B[(size_t)ka * SEQ + tn + l]; // B 4x16 tile: lane col
    b.y = B[(size_t)(ka + 1) * SEQ + tn + l];
    acc = __builtin_amdgcn_wmma_f32_16x16x4_f32(
        false, a, false, b, (short)0, acc, false, false);
  }
  int row0 = tm + half * 8;           // C/D layout: VGPR r -> rows r / r+8
  int col = tn + l;
#pragma unroll
  for (int r = 0; r < 8; ++r) C[(size_t)(row0 + r) * SEQ + col] = acc[r];
}

// --------------------------- per-step secular solve + output ----------------
__global__ __launch_bounds__(64) void k_secular(const float* __restrict__ ws,
                                                float* __restrict__ out) {
  __shared__ float dv[NB], w2[NB], lam[NB];
  __shared__ float trace_s, sumw_s, norm_s;
  int t = blockIdx.x;
  int j = threadIdx.x;
  dv[j] = ws[OFF_D + j];
  float z = ws[OFF_Z + (size_t)j * SEQ + t];
  w2[j] = z * z;
  __syncthreads();
  if (j == 0) {
    float msq = ws[OFF_MSQ + t];
    float s = 0.f, sd = 0.f;
    for (int i = 1; i < NB; ++i) s += w2[i];
    for (int i = 0; i < NB; ++i) sd += dv[i];
    float w0 = msq - s; if (w0 < 0.f) w0 = 0.f;  // aggregated null-space weight
    w2[0] = w0;
    trace_s = sd + msq;                           // tr(s^T s)/64 in closed form
    sumw_s = msq;
  }
  __syncthreads();
  // root j of 1 + sum w2_i/(d_i - x) in (d_j, d_{j+1}); f is increasing there
  float lo = dv[j];
  float hi = (j < NB - 1) ? dv[j + 1] : dv[NB - 1] + sumw_s + 1e-12f;
  for (int it = 0; it < 60; ++it) {
    float mid = 0.5f * (lo + hi);
    float f = 1.f;
    for (int i = 0; i < NB; ++i) f += w2[i] / (dv[i] - mid);
    if (f > 0.f) hi = mid; else lo = mid;
  }
  float ev = 0.5f * (lo + hi) / trace_s;
  if (ev < EPSV) ev = EPSV;                       // clip
  lam[j] = ev;
  __syncthreads();
  if (j == 0) {
    float s = 192.f * EPSV;                       // 192 exact-zero eigenvalues
    for (int i = 0; i < NB; ++i) s += lam[i];
    norm_s = 1.f / s;
  }
  __syncthreads();
  float inv = norm_s;
  for (int k = j; k < 192; k += 64) out[(size_t)t * DD + k] = EPSV * inv;
  out[(size_t)t * DD + 192 + j] = lam[j] * inv;   // ascending order preserved
}

// --------------------------- launcher ---------------------------------------
extern "C" void kernel_launch(void* const* d_in, const int* in_sizes, int n_in,
                              void* d_out, int out_size, void* d_ws, size_t ws_size,
                              hipStream_t stream) {
  const int* tokens = (const int*)d_in[0];
  const float* embed = (const float*)d_in[1];
  const float* bubbles = (const float*)d_in[2];
  const float* mdecay = (const float*)d_in[3];
  float* ws = (float*)d_ws;
  float* out = (float*)d_out;

  k_setup<<<dim3(1), dim3(256), 0, stream>>>(bubbles, mdecay, ws);
  k_gram<<<dim3(16), dim3(256), 0, stream>>>(ws);
  k_jacobi<<<dim3(1), dim3(256), 0, stream>>>(ws);
  k_scan<<<dim3(1), dim3(256), 0, stream>>>(tokens, embed, ws);
  k_gemm_wmma<256, 256><<<dim3(32), dim3(256), 0, stream>>>(ws + OFF_B0, ws + OFF_M, ws + OFF_Y);
  k_gemm_wmma<64, 64><<<dim3(32), dim3(256), 0, stream>>>(ws + OFF_A2, ws + OFF_Y, ws + OFF_Z);
  k_secular<<<dim3(SEQ), dim3(64), 0, stream>>>(ws, out);
}